// ANIAEV_26955214749809
// MI455X (gfx1250) — compile-verified
//
#include <hip/hip_runtime.h>
#include <stdint.h>

#define BLOCK 256

// ---- compile-time constants from the reference problem ----
#define NSP        4
#define RDIV       16
#define ADIV       4
#define ASEC       4
#define OUT_COLS   224                       // 4*16 radial + 10*16 angular
#define ANG_BASE   64
#define RAD_ETA    16.0f
#define RSTART     0.8f
#define RSTEP      0.275f                    // (5.2-0.8)/16
#define ASTART     0.8f
#define ASTEP      0.675f                    // (3.5-0.8)/4
#define SQRT8      2.8284271247461903f
#define DIST_SCALE ((float)(1.4142135623730951))            // 0.5*sqrt(8)
#define SW_SCALE   ((float)(1.4142135623730951 / 65536.0))  // sqrt(2*0.5^32)
#define ZSTART     0.39269908169872414f      // pi/8
#define ZSTEP      0.7853981633974483f       // pi/4

// CDNA5 async-to-LDS path (ASYNCcnt), guarded so the file compiles either way.
#if defined(__gfx1250__) && __has_builtin(__builtin_amdgcn_global_load_async_to_lds_b32) && __has_builtin(__builtin_amdgcn_s_wait_asynccnt)
#define ANI_ASYNC 1
#else
#define ANI_ASYNC 0
#endif

#if ANI_ASYNC
typedef __attribute__((address_space(1))) int gint_t;   // global (cuda_device) AS
typedef __attribute__((address_space(3))) int lint_t;   // LDS (shared) AS
#endif

extern "C" __global__ void __launch_bounds__(BLOCK)
ani_zero(float* __restrict__ out, int n)
{
    int i = blockIdx.x * BLOCK + threadIdx.x;
    if (i < n) out[i] = 0.0f;
}

// Pack per-angular-edge record: {sqrt(2)*d, sw*sqrt(2*0.5^32), species, pad}
extern "C" __global__ void __launch_bounds__(BLOCK)
ani_prep(const float* __restrict__ ang_d, const float* __restrict__ ang_sw,
         const int* __restrict__ ang_edge_dst, const int* __restrict__ species,
         float4* __restrict__ rec, int nea)
{
    int e = blockIdx.x * BLOCK + threadIdx.x;
    if (e >= nea) return;
    float d  = DIST_SCALE * ang_d[e];
    float w  = SW_SCALE   * ang_sw[e];
    int   sp = species[ang_edge_dst[e]];
    rec[e] = make_float4(d, w, __int_as_float(sp), 0.0f);
}

__device__ __forceinline__ void ani_radial_body(
    float d, float swv, int s, int sp,
    const int* __restrict__ species, float* __restrict__ out)
{
    float w = 0.25f * swv;
    float* base = out + (size_t)s * OUT_COLS + sp * RDIV;
#pragma unroll
    for (int j = 0; j < RDIV; ++j) {
        float x = d - (RSTART + RSTEP * (float)j);
        atomicAdd(base + j, w * __expf(-RAD_ETA * x * x));
    }
}

extern "C" __global__ void __launch_bounds__(BLOCK)
ani_radial(const float* __restrict__ dist, const float* __restrict__ sw,
           const int* __restrict__ esrc, const int* __restrict__ edst,
           const int* __restrict__ species, float* __restrict__ out, int ner)
{
    int e = blockIdx.x * BLOCK + threadIdx.x;
#if ANI_ASYNC
    // Stage the four coalesced streams through the CDNA5 async DMA path
    // (ASYNCcnt) so the fills overlap with kernel front-end/address setup.
    __shared__ float s_d[BLOCK];
    __shared__ float s_w[BLOCK];
    __shared__ int   s_s[BLOCK];
    __shared__ int   s_t[BLOCK];
    int tid = threadIdx.x;
    if (e < ner) {
        __builtin_amdgcn_global_load_async_to_lds_b32((gint_t*)(dist + e), (lint_t*)&s_d[tid], 0, 0);
        __builtin_amdgcn_global_load_async_to_lds_b32((gint_t*)(sw   + e), (lint_t*)&s_w[tid], 0, 0);
        __builtin_amdgcn_global_load_async_to_lds_b32((gint_t*)(esrc + e), (lint_t*)&s_s[tid], 0, 0);
        __builtin_amdgcn_global_load_async_to_lds_b32((gint_t*)(edst + e), (lint_t*)&s_t[tid], 0, 0);
        __builtin_amdgcn_s_wait_asynccnt(0);
    }
    if (e >= ner) return;
    ani_radial_body(s_d[tid], s_w[tid], s_s[tid], species[s_t[tid]], species, out);
#else
    if (e >= ner) return;
    ani_radial_body(dist[e], sw[e], esrc[e], species[edst[e]], species, out);
#endif
}

__device__ __forceinline__ void ani_angular_body(
    float ang, int c, int es, int ed,
    const float4* __restrict__ rec, float* __restrict__ out)
{
    float4 rs = rec[es];
    float4 rd = rec[ed];
    float f1  = rs.y * rd.y;            // includes 2*0.5^32 scale
    float d12 = rs.x + rd.x;
    int si = __float_as_int(rs.z), sj = __float_as_int(rd.z);
    int i = min(si, sj), j = max(si, sj);
    int pair = i * NSP - ((i * (i - 1)) >> 1) + (j - i);   // triu(4) linear index
    float* base = out + (size_t)c * OUT_COLS + ANG_BASE + pair * (ADIV * ASEC);

    float fz[ASEC];
#pragma unroll
    for (int z = 0; z < ASEC; ++z) {
        float x  = 1.0f + __cosf(ang - (ZSTART + ZSTEP * (float)z));
        float x2 = x * x, x4 = x2 * x2, x8 = x4 * x4, x16 = x8 * x8;
        fz[z] = f1 * (x16 * x16);       // (1+cos)^32, max 2^32 * 2^-31 = 2
    }
#pragma unroll
    for (int a = 0; a < ADIV; ++a) {
        float y  = d12 - SQRT8 * (ASTART + ASTEP * (float)a);
        float fa = __expf(-y * y);
#pragma unroll
        for (int z = 0; z < ASEC; ++z)
            atomicAdd(base + a * ASEC + z, fa * fz[z]);
    }
}

extern "C" __global__ void __launch_bounds__(BLOCK)
ani_angular(const float* __restrict__ angles, const int* __restrict__ cat,
            const int* __restrict__ asrc, const int* __restrict__ adst,
            const float4* __restrict__ rec, float* __restrict__ out, int np)
{
    int p = blockIdx.x * BLOCK + threadIdx.x;
#if ANI_ASYNC
    __shared__ float s_ang[BLOCK];
    __shared__ int   s_cat[BLOCK];
    __shared__ int   s_src[BLOCK];
    __shared__ int   s_dst[BLOCK];
    int tid = threadIdx.x;
    if (p < np) {
        __builtin_amdgcn_global_load_async_to_lds_b32((gint_t*)(angles + p), (lint_t*)&s_ang[tid], 0, 0);
        __builtin_amdgcn_global_load_async_to_lds_b32((gint_t*)(cat    + p), (lint_t*)&s_cat[tid], 0, 0);
        __builtin_amdgcn_global_load_async_to_lds_b32((gint_t*)(asrc   + p), (lint_t*)&s_src[tid], 0, 0);
        __builtin_amdgcn_global_load_async_to_lds_b32((gint_t*)(adst   + p), (lint_t*)&s_dst[tid], 0, 0);
        __builtin_amdgcn_s_wait_asynccnt(0);
    }
    if (p >= np) return;
    ani_angular_body(s_ang[tid], s_cat[tid], s_src[tid], s_dst[tid], rec, out);
#else
    if (p >= np) return;
    ani_angular_body(angles[p], cat[p], asrc[p], adst[p], rec, out);
#endif
}

// Fallback when the workspace cannot hold the packed EA records: do the
// double-indirect gathers inline (8 small gathers instead of 2x16B).
extern "C" __global__ void __launch_bounds__(BLOCK)
ani_angular_noprep(const float* __restrict__ angles, const int* __restrict__ cat,
                   const int* __restrict__ asrc, const int* __restrict__ adst,
                   const float* __restrict__ ang_d, const float* __restrict__ ang_sw,
                   const int* __restrict__ ang_edge_dst, const int* __restrict__ species,
                   float* __restrict__ out, int np)
{
    int p = blockIdx.x * BLOCK + threadIdx.x;
    if (p >= np) return;
    int es = asrc[p], ed = adst[p];
    float ds = DIST_SCALE * ang_d[es], dd = DIST_SCALE * ang_d[ed];
    float ws = SW_SCALE * ang_sw[es],  wd = SW_SCALE * ang_sw[ed];
    int sps = species[ang_edge_dst[es]];
    int spd = species[ang_edge_dst[ed]];
    float ang = angles[p];
    int c = cat[p];

    float f1  = ws * wd;
    float d12 = ds + dd;
    int i = min(sps, spd), j = max(sps, spd);
    int pair = i * NSP - ((i * (i - 1)) >> 1) + (j - i);
    float* base = out + (size_t)c * OUT_COLS + ANG_BASE + pair * (ADIV * ASEC);

    float fz[ASEC];
#pragma unroll
    for (int z = 0; z < ASEC; ++z) {
        float x  = 1.0f + __cosf(ang - (ZSTART + ZSTEP * (float)z));
        float x2 = x * x, x4 = x2 * x2, x8 = x4 * x4, x16 = x8 * x8;
        fz[z] = f1 * (x16 * x16);
    }
#pragma unroll
    for (int a = 0; a < ADIV; ++a) {
        float y  = d12 - SQRT8 * (ASTART + ASTEP * (float)a);
        float fa = __expf(-y * y);
#pragma unroll
        for (int z = 0; z < ASEC; ++z)
            atomicAdd(base + a * ASEC + z, fa * fz[z]);
    }
}

extern "C" void kernel_launch(void* const* d_in, const int* in_sizes, int n_in,
                              void* d_out, int out_size, void* d_ws, size_t ws_size,
                              hipStream_t stream)
{
    const int*   species = (const int*)  d_in[0];
    const float* rad_d   = (const float*)d_in[1];
    const float* rad_sw  = (const float*)d_in[2];
    const int*   esrc    = (const int*)  d_in[3];
    const int*   edst    = (const int*)  d_in[4];
    const float* ang_d   = (const float*)d_in[5];
    const float* ang_sw  = (const float*)d_in[6];
    const float* angles  = (const float*)d_in[7];
    const int*   cat     = (const int*)  d_in[8];
    const int*   asrc    = (const int*)  d_in[9];
    const int*   adst    = (const int*)  d_in[10];
    const int*   aedst   = (const int*)  d_in[11];

    const int ner = in_sizes[1];   // radial edges
    const int nea = in_sizes[5];   // angular edges
    const int np  = in_sizes[7];   // angle triples

    float* out = (float*)d_out;

    // Zero accumulator (harness poisons d_out; atomics accumulate).
    {
        int blocks = (out_size + BLOCK - 1) / BLOCK;
        hipLaunchKernelGGL(ani_zero, dim3(blocks), dim3(BLOCK), 0, stream, out, out_size);
    }

    // Radial pass (independent of prep).
    {
        int blocks = (ner + BLOCK - 1) / BLOCK;
        hipLaunchKernelGGL(ani_radial, dim3(blocks), dim3(BLOCK), 0, stream,
                           rad_d, rad_sw, esrc, edst, species, out, ner);
    }

    // Angular pass: prefer packed per-edge records in workspace (L2-resident).
    size_t need = (size_t)nea * sizeof(float4);
    if (ws_size >= need) {
        float4* rec = (float4*)d_ws;
        int pblocks = (nea + BLOCK - 1) / BLOCK;
        hipLaunchKernelGGL(ani_prep, dim3(pblocks), dim3(BLOCK), 0, stream,
                           ang_d, ang_sw, aedst, species, rec, nea);
        int ablocks = (np + BLOCK - 1) / BLOCK;
        hipLaunchKernelGGL(ani_angular, dim3(ablocks), dim3(BLOCK), 0, stream,
                           angles, cat, asrc, adst, rec, out, np);
    } else {
        int ablocks = (np + BLOCK - 1) / BLOCK;
        hipLaunchKernelGGL(ani_angular_noprep, dim3(ablocks), dim3(BLOCK), 0, stream,
                           angles, cat, asrc, adst, ang_d, ang_sw, aedst, species, out, np);
    }
}